// SelfAttention_17970143166886
// MI455X (gfx1250) — compile-verified
//
#include <hip/hip_runtime.h>
#include <hip/hip_bf16.h>
#include <math.h>

// Problem constants (match reference)
#define BATCH   8
#define CIN_    256
#define NPIX    4096          // H*W
#define HEADS_  4
#define HC_     64
#define HIDDEN_ 256           // HEADS*HC
#define QKV_O   768           // 3*HIDDEN
#define COUT_   256
#define EPS_    1e-5f

typedef __attribute__((ext_vector_type(2))) float v2f;
typedef __attribute__((ext_vector_type(8))) float v8f;

// ---------------------------------------------------------------------------
// Generic fp32 WMMA tile GEMM:  C[m,n] = sum_k A[m,k]*B[k,n] (+ bias[m])
//   A element (m,k) at A[m*sa_m + k*sa_k]
//   B element (k,n) at B[k*sb_k + n*sb_n]
//   C row-major with leading dim ldc
// One wave computes a 16 x (16*NT) tile via V_WMMA_F32_16X16X4_F32, reusing
// the A fragment across the NT accumulators. EXEC is all-1s (blockDim==32,
// uniform control flow) per WMMA restrictions.
//
// VGPR layouts (CDNA5 ISA 7.12.2):
//   A 16x4 : lanes 0-15 -> K={0,1}, lanes 16-31 -> K={2,3}; M = lane%16
//   B 4x16 : lanes 0-15 -> K={0,1}, lanes 16-31 -> K={2,3}; N = lane%16
//   D 16x16: VGPR r -> row r + 8*(lane/16), col = lane%16
// ---------------------------------------------------------------------------
template <int NT, bool HB>
__device__ __forceinline__ void wmma_tile_f32(
    const float* __restrict__ A, const float* __restrict__ B,
    float* __restrict__ C, const float* __restrict__ bias,
    int tm, int tn, int K,
    long sa_m, long sa_k, long sb_k, long sb_n, long ldc)
{
    const int lane = threadIdx.x & 31;
    const int half = lane >> 4;     // 0 or 1
    const int l16  = lane & 15;

    const float* Ap = A + (long)(tm * 16 + l16) * sa_m;
    const float* Bp[NT];
    #pragma unroll
    for (int t = 0; t < NT; ++t)
        Bp[t] = B + (long)(tn * 16 * NT + t * 16 + l16) * sb_n;

    v8f acc[NT];
    #pragma unroll
    for (int t = 0; t < NT; ++t) acc[t] = (v8f){};

    #pragma unroll 4
    for (int kk = 0; kk < K; kk += 4) {
        const long ka = (long)(kk + 2 * half);
        v2f a;
        a.x = Ap[ka * sa_k];
        a.y = Ap[(ka + 1) * sa_k];
        #pragma unroll
        for (int t = 0; t < NT; ++t) {
            v2f bf;
            bf.x = Bp[t][ka * sb_k];
            bf.y = Bp[t][(ka + 1) * sb_k];
            acc[t] = __builtin_amdgcn_wmma_f32_16x16x4_f32(
                false, a, false, bf, (short)0, acc[t], false, false);
        }
    }

    #pragma unroll
    for (int r = 0; r < 8; ++r) {
        const long row = (long)(tm * 16 + r + 8 * half);
        float bv = 0.f;
        if (HB) bv = bias[row];
        #pragma unroll
        for (int t = 0; t < NT; ++t) {
            const long col = (long)(tn * 16 * NT + t * 16 + l16);
            C[row * ldc + col] = acc[t][r] + bv;
        }
    }
}

// ---------------------------------------------------------------------------
// K1: qkv[b,o,n] = sum_c Wqkv[o,c] * x[b,c,n] + bqkv[o]
//     M=768, K=256, N=4096 per batch. grid=(48*128, 8), block=32
// ---------------------------------------------------------------------------
__global__ void qkv_gemm_kernel(const float* __restrict__ x,
                                const float* __restrict__ Wqkv,
                                const float* __restrict__ bqkv,
                                float* __restrict__ qkv)
{
    const int b  = blockIdx.y;
    const int tn = blockIdx.x & 127;          // NPIX/32 = 128 tile-pairs
    const int tm = blockIdx.x >> 7;           // 0..47
    wmma_tile_f32<2, true>(Wqkv,
                  x   + (long)b * CIN_  * NPIX,
                  qkv + (long)b * QKV_O * NPIX,
                  bqkv,
                  tm, tn, CIN_,
                  /*sa_m*/ CIN_, /*sa_k*/ 1,
                  /*sb_k*/ NPIX, /*sb_n*/ 1,
                  /*ldc*/  NPIX);
}

// ---------------------------------------------------------------------------
// K2: softmax over n for the K part of qkv (o in [256,512)), in place.
//     grid = 8*256 rows, block=256
// ---------------------------------------------------------------------------
__global__ void softmax_rows_kernel(float* __restrict__ qkv)
{
    const int b = blockIdx.x >> 8;
    const int r = blockIdx.x & 255;
    float* p = qkv + ((long)b * QKV_O + 256 + r) * (long)NPIX;
    const int tid = threadIdx.x;
    __shared__ float sh[256];

    float m = -INFINITY;
    for (int i = tid; i < NPIX; i += 256) m = fmaxf(m, p[i]);
    sh[tid] = m; __syncthreads();
    for (int s = 128; s > 0; s >>= 1) {
        if (tid < s) sh[tid] = fmaxf(sh[tid], sh[tid + s]);
        __syncthreads();
    }
    m = sh[0]; __syncthreads();

    float sum = 0.f;
    for (int i = tid; i < NPIX; i += 256) {
        float e = __expf(p[i] - m);
        p[i] = e;
        sum += e;
    }
    sh[tid] = sum; __syncthreads();
    for (int s = 128; s > 0; s >>= 1) {
        if (tid < s) sh[tid] += sh[tid + s];
        __syncthreads();
    }
    const float inv = 1.0f / sh[0];
    for (int i = tid; i < NPIX; i += 256) p[i] *= inv;
}

// ---------------------------------------------------------------------------
// K3: ctx[b,h,d,e] = sum_n k[b,h,d,n] * v[b,h,e,n]
//     per (b,h): M=64 (d), N=64 (e), K=4096 (n)
//     A = K-part rows (row-major over n). B(k=n, n=e) = v[e*NPIX + n].
//     grid=(8, 32), block=32
// ---------------------------------------------------------------------------
__global__ void ctx_gemm_kernel(const float* __restrict__ qkv,
                                float* __restrict__ ctx)
{
    const int z = blockIdx.y;         // b*4 + h
    const int b = z >> 2, h = z & 3;
    const float* Kp = qkv + ((long)b * QKV_O + 256 + h * HC_) * (long)NPIX;
    const float* Vp = qkv + ((long)b * QKV_O + 512 + h * HC_) * (long)NPIX;
    float* Cp = ctx + (long)z * HC_ * HC_;
    const int tm = blockIdx.x >> 1;   // 0..3
    const int tn = blockIdx.x & 1;    // 0..1 (pairs of 16 -> 32 cols)
    wmma_tile_f32<2, false>(Kp, Vp, Cp, nullptr,
                  tm, tn, NPIX,
                  /*sa_m*/ NPIX, /*sa_k*/ 1,
                  /*sb_k*/ 1,    /*sb_n*/ NPIX,
                  /*ldc*/  HC_);
}

// ---------------------------------------------------------------------------
// K4: out[b,h,e,n] = sum_d ctx[b,h,d,e] * q[b,h,d,n]
//     per (b,h): M=64 (e), K=64 (d), N=4096.
//     A(m=e,k=d) = ctx[d*64 + e] (col-major).  Result overwrites the (dead)
//     V region of qkv -> becomes the "out" (B,256,N) tensor for K5.
//     grid=(4*128, 32), block=32
// ---------------------------------------------------------------------------
__global__ void attnout_gemm_kernel(float* __restrict__ qkv,
                                    const float* __restrict__ ctx)
{
    const int z = blockIdx.y;
    const int b = z >> 2, h = z & 3;
    const float* Qp = qkv + ((long)b * QKV_O + h * HC_) * (long)NPIX;        // q
    const float* Ax = ctx + (long)z * HC_ * HC_;
    float* Op = qkv + ((long)b * QKV_O + 512 + h * HC_) * (long)NPIX;        // out
    const int tm = blockIdx.x >> 7;   // 0..3
    const int tn = blockIdx.x & 127;  // 0..127
    wmma_tile_f32<2, false>(Ax, Qp, Op, nullptr,
                  tm, tn, HC_,
                  /*sa_m*/ 1,    /*sa_k*/ HC_,
                  /*sb_k*/ NPIX, /*sb_n*/ 1,
                  /*ldc*/  NPIX);
}

// ---------------------------------------------------------------------------
// K5: y[b,co,n] = sum_c Wout[co,c] * out[b,c,n] + bout[co]   (y lives in d_out)
//     M=256, K=256, N=4096 per batch. grid=(16*128, 8), block=32
// ---------------------------------------------------------------------------
__global__ void proj_gemm_kernel(const float* __restrict__ qkv,
                                 const float* __restrict__ Wout,
                                 const float* __restrict__ bout,
                                 float* __restrict__ y)
{
    const int b  = blockIdx.y;
    const int tn = blockIdx.x & 127;
    const int tm = blockIdx.x >> 7;   // 0..15
    const float* Xp = qkv + ((long)b * QKV_O + 512) * (long)NPIX;  // out region
    wmma_tile_f32<2, true>(Wout, Xp, y + (long)b * COUT_ * NPIX, bout,
                  tm, tn, HIDDEN_,
                  /*sa_m*/ HIDDEN_, /*sa_k*/ 1,
                  /*sb_k*/ NPIX,    /*sb_n*/ 1,
                  /*ldc*/  NPIX);
}

// ---------------------------------------------------------------------------
// K6a: per-batch partial (sum, sumsq). grid=(256, 8), block=256
// ---------------------------------------------------------------------------
__global__ void ln_partial_kernel(const float* __restrict__ y,
                                  float* __restrict__ partial)
{
    const long base = (long)blockIdx.y * COUT_ * NPIX + (long)blockIdx.x * 4096;
    const int tid = threadIdx.x;
    float s = 0.f, s2 = 0.f;
    for (int i = tid; i < 4096; i += 256) {
        float v = y[base + i];
        s += v; s2 += v * v;
    }
    __shared__ float sh[256], sh2[256];
    sh[tid] = s; sh2[tid] = s2; __syncthreads();
    for (int st = 128; st > 0; st >>= 1) {
        if (tid < st) { sh[tid] += sh[tid + st]; sh2[tid] += sh2[tid + st]; }
        __syncthreads();
    }
    if (tid == 0) {
        const long o = ((long)blockIdx.y * 256 + blockIdx.x) * 2;
        partial[o]     = sh[0];
        partial[o + 1] = sh2[0];
    }
}

// K6b: reduce 256 partials per batch -> mean, inv_std. grid=8, block=256
__global__ void ln_finalize_kernel(const float* __restrict__ partial,
                                   float* __restrict__ stats)
{
    const int b = blockIdx.x;
    const int tid = threadIdx.x;
    __shared__ float sh[256], sh2[256];
    const long o = ((long)b * 256 + tid) * 2;
    sh[tid] = partial[o]; sh2[tid] = partial[o + 1]; __syncthreads();
    for (int st = 128; st > 0; st >>= 1) {
        if (tid < st) { sh[tid] += sh[tid + st]; sh2[tid] += sh2[tid + st]; }
        __syncthreads();
    }
    if (tid == 0) {
        const float cnt  = (float)COUT_ * (float)NPIX;
        const float mean = sh[0] / cnt;
        const float var  = sh2[0] / cnt - mean * mean;
        stats[b * 2]     = mean;
        stats[b * 2 + 1] = rsqrtf(var + EPS_);
    }
}

// K7: yn = (y-mean)*invstd*gamma+beta; out = tanh(relu(yn)). In-place on d_out.
__global__ void ln_apply_kernel(float* __restrict__ y,
                                const float* __restrict__ stats,
                                const float* __restrict__ gamma,
                                const float* __restrict__ beta)
{
    const long idx = (long)blockIdx.x * 256 + threadIdx.x;
    const int b = (int)(idx >> 20);           // COUT_*NPIX = 2^20
    const int c = (int)(idx >> 12) & 255;     // NPIX = 2^12
    const float mean = stats[b * 2];
    const float inv  = stats[b * 2 + 1];
    float v = (y[idx] - mean) * inv * gamma[c] + beta[c];
    v = (v > 0.f) ? tanhf(v) : 0.f;
    y[idx] = v;
}

// ---------------------------------------------------------------------------
extern "C" void kernel_launch(void* const* d_in, const int* in_sizes, int n_in,
                              void* d_out, int out_size, void* d_ws, size_t ws_size,
                              hipStream_t stream)
{
    const float* x     = (const float*)d_in[0];
    const float* Wqkv  = (const float*)d_in[1];
    const float* bqkv  = (const float*)d_in[2];
    const float* Wout  = (const float*)d_in[3];
    const float* bout  = (const float*)d_in[4];
    const float* gamma = (const float*)d_in[5];
    const float* beta  = (const float*)d_in[6];
    float* y = (float*)d_out;                       // y lives in d_out

    float* ws_f = (float*)d_ws;
    const long QKV_ELEMS = (long)BATCH * QKV_O * NPIX;        // 25,165,824
    float* qkv     = ws_f;
    float* ctx     = ws_f + QKV_ELEMS;                        // 131,072 floats
    float* partial = ctx + (long)BATCH * HEADS_ * HC_ * HC_;  // 4,096 floats
    float* stats   = partial + BATCH * 256 * 2;               // 16 floats

    // 1) QKV projection
    qkv_gemm_kernel<<<dim3(48 * 128, BATCH), 32, 0, stream>>>(x, Wqkv, bqkv, qkv);
    // 2) softmax over spatial dim for K
    softmax_rows_kernel<<<BATCH * 256, 256, 0, stream>>>(qkv);
    // 3) ctx = K . V^T  per (b,h)
    ctx_gemm_kernel<<<dim3(8, BATCH * HEADS_), 32, 0, stream>>>(qkv, ctx);
    // 4) out = ctx^T . Q  (overwrites V region of qkv)
    attnout_gemm_kernel<<<dim3(4 * 128, BATCH * HEADS_), 32, 0, stream>>>(qkv, ctx);
    // 5) output projection -> y (in d_out)
    proj_gemm_kernel<<<dim3(16 * 128, BATCH), 32, 0, stream>>>(qkv, Wout, bout, y);
    // 6) per-batch LayerNorm stats (deterministic two-stage reduction)
    ln_partial_kernel<<<dim3(256, BATCH), 256, 0, stream>>>(y, partial);
    ln_finalize_kernel<<<BATCH, 256, 0, stream>>>(partial, stats);
    // 7) normalize + affine + relu + tanh, in place
    ln_apply_kernel<<<(BATCH * COUT_ * NPIX) / 256, 256, 0, stream>>>(y, stats, gamma, beta);
}